// NaiveSVConv_62989990363678
// MI455X (gfx1250) — compile-verified
//
#include <hip/hip_runtime.h>

// Problem constants (from reference)
#define B_   32      // batch
#define C_   8       // in channels
#define O_   32      // out channels
#define P_   72      // C*K*K contraction dim (= 18 * 4, exact for f32 WMMA)
#define FSZ  64      // image side
#define F2   4096    // pixels
#define TF   8       // pixels per block
#define PSTR 74      // padded P stride in LDS (even -> 8B aligned fragments, stride%64=10 -> conflict-free)
#define YSTR 9       // padded TF stride for Y staging

typedef __attribute__((ext_vector_type(2))) float v2f;
typedef __attribute__((ext_vector_type(8))) float v8f;

#define GLOBAL_AS __attribute__((address_space(1)))
#define LDS_AS    __attribute__((address_space(3)))

// gfx1250 async global->LDS path (guarded so a missing builtin degrades to the
// synchronous staging path instead of a compile failure).
#if __has_builtin(__builtin_amdgcn_global_load_async_to_lds_b32)
#define HAVE_ASYNC 1
#else
#define HAVE_ASYNC 0
#endif

static __device__ __forceinline__ GLOBAL_AS int* as_gbl(const void* p) {
    // AMDGPU global pointers are the plain 64-bit VA.
    return (GLOBAL_AS int*)(unsigned long long)p;
}
static __device__ __forceinline__ LDS_AS int* as_lds(void* p) {
    // Generic LDS pointer: low 32 bits are the wave-relative LDS byte address.
    return (LDS_AS int*)(unsigned int)(unsigned long long)p;
}

static __device__ __forceinline__ void wait_async0() {
#if __has_builtin(__builtin_amdgcn_s_wait_asynccnt)
    __builtin_amdgcn_s_wait_asynccnt(0);
#else
    asm volatile("s_wait_asynccnt 0" ::: "memory");
#endif
}

// ws float layout: [0..31]=sum  [32..63]=sumsq  [64..95]=scale  [96..127]=shift

__global__ void zero_ws_kernel(float* ws) {
    if (threadIdx.x < 64) ws[threadIdx.x] = 0.0f;
}

__global__ __launch_bounds__(256, 1)
void svconv_wmma_kernel(const float* __restrict__ x,
                        const float* __restrict__ w,
                        const float* __restrict__ bias,
                        float* __restrict__ out,
                        float* __restrict__ ws) {
    extern __shared__ float smem[];
    float* Wl  = smem;                       // [TF][O_][PSTR]  weights, transposed to pixel-major
    float* Al  = Wl + TF * O_ * PSTR;        // [TF][B_][PSTR]  im2col patches
    float* Yl  = Al + TF * B_ * PSTR;        // [B_][O_][YSTR]  result staging
    float* red = Yl + B_ * O_ * YSTR;        // [64] per-channel sum / sumsq

    const int tid = threadIdx.x;
    const int f0  = blockIdx.x * TF;

    if (tid < 64) red[tid] = 0.0f;

    // ---- stage weight slice: w[(o*P+p)*F2 + f] -> Wl[(fi*O+o)*PSTR + p] ----
    // Async DMA does the scatter-transpose directly into LDS (no VGPR round-trip).
    for (int idx = tid; idx < TF * O_ * P_; idx += 256) {
        int fi   = idx & (TF - 1);
        int rest = idx >> 3;
        int p    = rest % P_;
        int o    = rest / P_;
        const float* src = &w[(o * P_ + p) * F2 + f0 + fi];
        float*       dst = &Wl[(fi * O_ + o) * PSTR + p];
#if HAVE_ASYNC
        __builtin_amdgcn_global_load_async_to_lds_b32(as_gbl(src), as_lds(dst), 0, 0);
#else
        *dst = *src;
#endif
    }

    // ---- stage im2col patches: Al[(fi*B+b)*PSTR + p] = x[b][c][h+kh-1][w+kw-1] ----
    for (int idx = tid; idx < TF * B_ * P_; idx += 256) {
        int fi   = idx & (TF - 1);
        int rest = idx >> 3;
        int p    = rest % P_;
        int b    = rest / P_;
        int c  = p / 9;
        int r  = p - c * 9;
        int kh = r / 3;
        int kw = r - kh * 3;
        int f  = f0 + fi;
        int h    = (f >> 6) + kh - 1;
        int wcol = (f & 63) + kw - 1;
        float* dst = &Al[(fi * B_ + b) * PSTR + p];
        bool inb = (h >= 0 && h < FSZ && wcol >= 0 && wcol < FSZ);
#if HAVE_ASYNC
        if (inb) {
            const float* src = &x[((b * C_ + c) * FSZ + h) * FSZ + wcol];
            __builtin_amdgcn_global_load_async_to_lds_b32(as_gbl(src), as_lds(dst), 0, 0);
        } else {
            *dst = 0.0f;
        }
#else
        *dst = inb ? x[((b * C_ + c) * FSZ + h) * FSZ + wcol] : 0.0f;
#endif
    }

#if HAVE_ASYNC
    wait_async0();       // drain this wave's ASYNCcnt; barrier publishes LDS to all waves
#endif
    __syncthreads();

    // ---- WMMA: per pixel a 32x32x72 GEMM as four 16x16 tiles, 18 k-steps of 4 ----
    const int wave = tid >> 5;
    const int lane = tid & 31;
    const int lrow = lane & 15;       // A: M (batch) | B: N (out chan)
    const int half = lane >> 4;       // K offset 0 / 2 within the k-step

    for (int j = 0; j < 4; ++j) {
        int jid = wave * 4 + j;           // 0..31 tile jobs
        int fi  = jid >> 2;               // pixel within block
        int q   = jid & 3;                // quadrant
        int m0  = (q & 1) << 4;           // batch tile base
        int n0  = (q >> 1) << 4;          // out-channel tile base

        v8f acc = {};
        const float* Abase = &Al[(fi * B_ + m0 + lrow) * PSTR + (half << 1)];
        const float* Bbase = &Wl[(fi * O_ + n0 + lrow) * PSTR + (half << 1)];
#pragma unroll
        for (int ks = 0; ks < 18; ++ks) {
            v2f a  = *(const v2f*)(Abase + 4 * ks);
            v2f bm = *(const v2f*)(Bbase + 4 * ks);
            // (neg_a, A, neg_b, B, c_mod, C, reuse_a, reuse_b)
            acc = __builtin_amdgcn_wmma_f32_16x16x4_f32(false, a, false, bm,
                                                        (short)0, acc, false, false);
        }
        // C/D layout: VGPR g -> (M = g + 8*half, N = lrow)
#pragma unroll
        for (int g = 0; g < 8; ++g) {
            int b = m0 + g + (half << 3);
            int o = n0 + lrow;
            Yl[(b * O_ + o) * YSTR + fi] = acc[g];
        }
    }

    __syncthreads();

    // ---- epilogue: bias add, f-contiguous global store, per-channel stats ----
    for (int r = 0; r < 4; ++r) {
        int pair = tid + 256 * r;         // 0..1023 (b,o) pairs
        int b = pair >> 5;
        int o = pair & 31;
        float s = 0.0f, s2 = 0.0f;
        float vals[TF];
#pragma unroll
        for (int fi = 0; fi < TF; ++fi) {
            float v = Yl[(b * O_ + o) * YSTR + fi] + bias[o * F2 + f0 + fi];
            vals[fi] = v;
            s  += v;
            s2 += v * v;
        }
        float4* dst = (float4*)&out[(b * O_ + o) * F2 + f0];
        dst[0] = make_float4(vals[0], vals[1], vals[2], vals[3]);
        dst[1] = make_float4(vals[4], vals[5], vals[6], vals[7]);
        atomicAdd(&red[o],      s);
        atomicAdd(&red[32 + o], s2);
    }

    __syncthreads();
    if (tid < 32) {
        atomicAdd(&ws[tid],      red[tid]);
        atomicAdd(&ws[32 + tid], red[32 + tid]);
    }
}

__global__ void bn_finalize_kernel(const float* __restrict__ gamma,
                                   const float* __restrict__ beta,
                                   float* __restrict__ ws) {
    int o = threadIdx.x;
    if (o < 32) {
        const float invN = 1.0f / (float)(B_ * F2);
        float mean = ws[o] * invN;
        float var  = ws[32 + o] * invN - mean * mean;
        float inv  = rsqrtf(var + 1e-5f);
        float sc   = gamma[o] * inv;
        ws[64 + o] = sc;
        ws[96 + o] = beta[o] - mean * sc;
    }
}

__global__ void bn_apply_kernel(float* __restrict__ out, const float* __restrict__ ws) {
    int idx = blockIdx.x * blockDim.x + threadIdx.x;   // float4 index
    int e   = idx << 2;
    int o   = (e >> 12) & 31;                          // (elem / F2) % O_
    float sc = ws[64 + o];
    float sh = ws[96 + o];
    float4* p = (float4*)out;
    float4 v = p[idx];
    v.x = v.x * sc + sh;  v.y = v.y * sc + sh;
    v.z = v.z * sc + sh;  v.w = v.w * sc + sh;
    p[idx] = v;
}

extern "C" void kernel_launch(void* const* d_in, const int* in_sizes, int n_in,
                              void* d_out, int out_size, void* d_ws, size_t ws_size,
                              hipStream_t stream) {
    (void)in_sizes; (void)n_in; (void)out_size; (void)ws_size;
    const float* x     = (const float*)d_in[0];
    const float* w     = (const float*)d_in[1];
    const float* bias  = (const float*)d_in[2];
    const float* gamma = (const float*)d_in[3];
    const float* beta  = (const float*)d_in[4];
    float* out = (float*)d_out;
    float* ws  = (float*)d_ws;

    zero_ws_kernel<<<1, 64, 0, stream>>>(ws);

    const size_t shbytes =
        (size_t)(TF * O_ * PSTR + TF * B_ * PSTR + B_ * O_ * YSTR + 64) * sizeof(float);
    svconv_wmma_kernel<<<F2 / TF, 256, shbytes, stream>>>(x, w, bias, out, ws);

    bn_finalize_kernel<<<1, 32, 0, stream>>>(gamma, beta, ws);

    const int n4 = (B_ * O_ * F2) / 4;
    bn_apply_kernel<<<n4 / 256, 256, 0, stream>>>(out, ws);
}